// DualSIMoptimizerTorch_36361193128541
// MI455X (gfx1250) — compile-verified
//
#include <hip/hip_runtime.h>
#include <math.h>
#include <stdint.h>

typedef __attribute__((ext_vector_type(2))) float v2f;
typedef __attribute__((ext_vector_type(8))) float v8f;

#define NDIM 2048
#define MDIM 64
#define LLAYERS 6
#define KC 16  // K-chunk staged through LDS per pipeline stage

// ---------------------------------------------------------------------------
// ds_load_2addr_b32: two strided dwords -> one consecutive VGPR pair (a v2f
// WMMA fragment) with zero register shuffles. Offsets are dword-unit
// immediates (ADJ=4 for b32).
// ---------------------------------------------------------------------------
template<int O0, int O1>
__device__ __forceinline__ v2f ds2(unsigned addr)
{
    v2f d;
    asm volatile("ds_load_2addr_b32 %0, %1 offset0:%c2 offset1:%c3"
                 : "=v"(d) : "v"(addr), "i"(O0), "i"(O1));
    return d;
}

// ---------------------------------------------------------------------------
// Head rotation: out = diag-phase applied to rows (rowmode=1) or cols (0).
// Used only for the two chain heads; every other diag(e^{i*phi}) factor is
// folded into the producing GEMM's epilogue.
// ---------------------------------------------------------------------------
__global__ __launch_bounds__(256)
void rotate_phase_kernel(const float* __restrict__ in, const float* __restrict__ ph,
                         float* __restrict__ out, int rows, int cols, int rowmode)
{
    int idx = blockIdx.x * blockDim.x + threadIdx.x;   // complex element index
    if (idx >= rows * cols) return;
    int r = idx / cols, c = idx - r * cols;
    float s, co;
    __sincosf(ph[rowmode ? r : c], &s, &co);
    float re = in[idx * 2], im = in[idx * 2 + 1];
    out[idx * 2]     = re * co - im * s;
    out[idx * 2 + 1] = re * s  + im * co;
}

// ---------------------------------------------------------------------------
// C[Mrows, Ncols] = A[Mrows, K] @ B[K, Ncols]      (complex64, row-major)
// epiMode: 0 = none, 1 = C row r *= e^{i*phase[r]}, 2 = C col c *= e^{i*phase[c]}
// Block = 128 threads (4 wave32), computes a 64x32 complex tile.
// LDS double-buffered; chunks staged with global_load_async_to_lds_b128
// (ASYNCcnt) so the fetch of chunk k+1 overlaps the WMMA work on chunk k.
// ---------------------------------------------------------------------------
__global__ __launch_bounds__(128)
void cgemm_async_wmma(const float* __restrict__ A, const float* __restrict__ B,
                      float* __restrict__ C, int Mrows, int K, int Ncols,
                      const float* __restrict__ epiPhase, int epiMode)
{
    constexpr int A_CPLX = 64 * KC;   // 1024 complex = 8 KB
    constexpr int B_CPLX = KC * 32;   //  512 complex = 4 KB
    __shared__ __align__(16) float2 sA[2][A_CPLX];
    __shared__ __align__(16) float2 sB[2][B_CPLX];

    const int tid  = threadIdx.x;
    const int lane = tid & 31;
    const int wave = tid >> 5;
    const int n0   = blockIdx.x * 32;
    const int m0   = blockIdx.y * 64;

    const float* Abase = A + (size_t)m0 * K * 2;

    // Stage one K-chunk (A: 64x16, B: 16x32 complex) into LDS buffer `buf`
    // via async copies: b128 = 2 complex per lane per issue, 6 issues/thread.
    auto stage = [&](int buf, int kc) {
        #pragma unroll
        for (int i = 0; i < 4; ++i) {
            int p  = tid + i * 128;          // complex-pair index 0..511
            int m  = p >> 3;                 // 8 pairs per 16-wide row
            int kp = (p & 7) << 1;
            uint64_t ga = (uint64_t)(uintptr_t)(Abase + ((size_t)m * K + kc + kp) * 2);
            unsigned la = (unsigned)(uintptr_t)&sA[buf][m * KC + kp];
            asm volatile("global_load_async_to_lds_b128 %0, %1, off"
                         :: "v"(la), "v"(ga) : "memory");
        }
        #pragma unroll
        for (int i = 0; i < 2; ++i) {
            int p  = tid + i * 128;          // complex-pair index 0..255
            int kk = p >> 4;                 // 16 pairs per 32-wide row
            int np = (p & 15) << 1;
            uint64_t gb = (uint64_t)(uintptr_t)(B + ((size_t)(kc + kk) * Ncols + n0 + np) * 2);
            unsigned lb = (unsigned)(uintptr_t)&sB[buf][kk * 32 + np];
            asm volatile("global_load_async_to_lds_b128 %0, %1, off"
                         :: "v"(lb), "v"(gb) : "memory");
        }
    };

    // Complex product via 4 real accumulators per N-tile:
    //   re = rr - ii ; im = ri + ir   (F32 WMMA NEG modifiers are C-only)
    v8f acc[8] = {};  // [rr0, ii0, ri0, ir0, rr1, ii1, ri1, ir1]

    const int mrow = (wave << 4) + (lane & 15);   // A row within the 64-row slab
    const int koff = (lane >> 4) << 1;            // lanes 16-31 carry K=+2 (32-bit A/B layout)
    const int ncol = lane & 15;

    // Per-wave LDS fragment base addresses (bytes), buffer 0.
    const unsigned aBase0 = (unsigned)(uintptr_t)&sA[0][0] + (unsigned)(mrow * KC * 8 + koff * 8);
    const unsigned bBase0 = (unsigned)(uintptr_t)&sB[0][0] + (unsigned)(ncol * 8 + koff * 256);

    stage(0, 0);
    asm volatile("s_wait_asynccnt 0" ::: "memory");
    __syncthreads();

    int cur = 0;
    for (int kc = 0; kc < K; kc += KC) {
        if (kc + KC < K) stage(cur ^ 1, kc + KC);   // prefetch next chunk (async)

        unsigned aAddr = aBase0 + (unsigned)cur * (A_CPLX * 8);
        unsigned bAddr = bBase0 + (unsigned)cur * (B_CPLX * 8);
        #pragma unroll
        for (int ks = 0; ks < KC / 4; ++ks) {
            v2f ar  = ds2<0, 2>(aAddr);     // A[m][kk..kk+1].re
            v2f ai  = ds2<1, 3>(aAddr);     // A[m][kk..kk+1].im
            v2f br0 = ds2<0, 64>(bAddr);    // B[kk..kk+1][n].re
            v2f bi0 = ds2<1, 65>(bAddr);    // B[kk..kk+1][n].im
            v2f br1 = ds2<32, 96>(bAddr);   // B[kk..kk+1][n+16].re
            v2f bi1 = ds2<33, 97>(bAddr);   // B[kk..kk+1][n+16].im
            // One wait for all six gathers; operand ties order loads->wait->WMMA.
            asm volatile("s_wait_dscnt 0"
                         : "+v"(ar), "+v"(ai), "+v"(br0), "+v"(bi0), "+v"(br1), "+v"(bi1));
            acc[0] = __builtin_amdgcn_wmma_f32_16x16x4_f32(false, ar, false, br0, (short)0, acc[0], false, false);
            acc[1] = __builtin_amdgcn_wmma_f32_16x16x4_f32(false, ai, false, bi0, (short)0, acc[1], false, false);
            acc[2] = __builtin_amdgcn_wmma_f32_16x16x4_f32(false, ar, false, bi0, (short)0, acc[2], false, false);
            acc[3] = __builtin_amdgcn_wmma_f32_16x16x4_f32(false, ai, false, br0, (short)0, acc[3], false, false);
            acc[4] = __builtin_amdgcn_wmma_f32_16x16x4_f32(false, ar, false, br1, (short)0, acc[4], false, false);
            acc[5] = __builtin_amdgcn_wmma_f32_16x16x4_f32(false, ai, false, bi1, (short)0, acc[5], false, false);
            acc[6] = __builtin_amdgcn_wmma_f32_16x16x4_f32(false, ar, false, bi1, (short)0, acc[6], false, false);
            acc[7] = __builtin_amdgcn_wmma_f32_16x16x4_f32(false, ai, false, br1, (short)0, acc[7], false, false);
            aAddr += 32;     // next K-step: +4 K -> +8 dwords in A row
            bAddr += 1024;   // next K-step: +4 B rows of 64 dwords
        }

        asm volatile("s_wait_asynccnt 0" ::: "memory");  // my chunk-(k+1) copies done
        __syncthreads();                                  // all waves done; swap buffers
        cur ^= 1;
    }

    // ---- Epilogue: combine complex parts, optional phase rotation, store ----
    const int rbase = m0 + (wave << 4) + ((lane >> 4) << 3);
    const int c0g = n0 + ncol, c1g = c0g + 16;
    float s0 = 0.f, co0 = 1.f, s1 = 0.f, co1 = 1.f;
    if (epiMode == 2) {
        __sincosf(epiPhase[c0g], &s0, &co0);
        __sincosf(epiPhase[c1g], &s1, &co1);
    }
    #pragma unroll
    for (int r = 0; r < 8; ++r) {
        int row = rbase + r;
        float re0 = acc[0][r] - acc[1][r];
        float im0 = acc[2][r] + acc[3][r];
        float re1 = acc[4][r] - acc[5][r];
        float im1 = acc[6][r] + acc[7][r];
        if (epiMode == 1) {
            float s, co;
            __sincosf(epiPhase[row], &s, &co);
            float t;
            t = re0 * co - im0 * s; im0 = re0 * s + im0 * co; re0 = t;
            t = re1 * co - im1 * s; im1 = re1 * s + im1 * co; re1 = t;
        } else if (epiMode == 2) {
            float t;
            t = re0 * co0 - im0 * s0; im0 = re0 * s0 + im0 * co0; re0 = t;
            t = re1 * co1 - im1 * s1; im1 = re1 * s1 + im1 * co1; re1 = t;
        }
        *(float2*)(C + ((size_t)row * Ncols + c0g) * 2) = make_float2(re0, im0);
        *(float2*)(C + ((size_t)row * Ncols + c1g) * 2) = make_float2(re1, im1);
    }
}

static inline void launch_cgemm(const float* A, const float* B, float* C,
                                int Mrows, int K, int Ncols,
                                const float* epiPhase, int epiMode, hipStream_t s)
{
    dim3 grid(Ncols / 32, Mrows / 64);
    cgemm_async_wmma<<<grid, 128, 0, s>>>(A, B, C, Mrows, K, Ncols, epiPhase, epiMode);
}

extern "C" void kernel_launch(void* const* d_in, const int* in_sizes, int n_in,
                              void* d_out, int out_size, void* d_ws, size_t ws_size,
                              hipStream_t stream)
{
    (void)in_sizes; (void)n_in; (void)out_size; (void)ws_size;
    const float* H     = (const float*)d_in[0];   // [N,N]     c64
    const float* W0_T  = (const float*)d_in[1];   // [N,M]     c64
    const float* WL_T  = (const float*)d_in[2];   // [N,N]     c64
    const float* WintT = (const float*)d_in[3];   // [L-1,N,N] c64
    const float* xi_T  = (const float*)d_in[4];   // [L,N]     f32
    const float* W0_R  = (const float*)d_in[5];   // [N,N]     c64
    const float* WL_R  = (const float*)d_in[6];   // [M,N]     c64
    const float* WintR = (const float*)d_in[7];   // [L-1,N,N] c64
    const float* xi_R  = (const float*)d_in[8];   // [L,N]     f32
    float* out = (float*)d_out;                   // [M,M] c64

    const size_t PANEL = (size_t)NDIM * MDIM * 2; // floats per 2048x64 complex panel (1 MB)
    float* bufA = (float*)d_ws;
    float* bufB = bufA + PANEL;
    float* bufC = bufB + PANEL;
    float* bufD = bufC + PANEL;

    const int PCNT = NDIM * MDIM;                 // complex elements per panel

    // ---- TX chain (right-to-left): GT = WL_T D5 W4 D4 ... W0 D0 W0_T ----
    rotate_phase_kernel<<<(PCNT + 255) / 256, 256, 0, stream>>>(
        W0_T, xi_T, bufA, NDIM, MDIM, 1);         // bufA = D0 . W0_T
    const float* g = bufA;
    float* gout = bufB;
    for (int l = 0; l < LLAYERS - 1; ++l) {
        // G <- D_{l+1} . (W_int_T[l] @ G)   (epilogue row rotation)
        launch_cgemm(WintT + (size_t)l * NDIM * NDIM * 2, g, gout,
                     NDIM, NDIM, MDIM, xi_T + (size_t)(l + 1) * NDIM, 1, stream);
        const float* t = g; g = gout; gout = (float*)t;
    }
    float* GT = gout;                              // ends in bufA
    launch_cgemm(WL_T, g, GT, NDIM, NDIM, MDIM, nullptr, 0, stream);

    // ---- RX chain (left-to-right): GR = WL_R D5 W4 D4 ... W0 D0 W0_R ----
    rotate_phase_kernel<<<(PCNT + 255) / 256, 256, 0, stream>>>(
        WL_R, xi_R + (size_t)(LLAYERS - 1) * NDIM, bufC, MDIM, NDIM, 0);  // bufC = WL_R . D5
    const float* x = bufC;
    float* xout = bufD;
    for (int l = LLAYERS - 2; l >= 0; --l) {
        // X <- (X @ W_int_R[l]) . D_l   (epilogue column rotation)
        launch_cgemm(x, WintR + (size_t)l * NDIM * NDIM * 2, xout,
                     MDIM, NDIM, NDIM, xi_R + (size_t)l * NDIM, 2, stream);
        const float* t = x; x = xout; xout = (float*)t;
    }
    float* GR = xout;                              // X_5 ended in bufD -> GR into bufC
    launch_cgemm(x, W0_R, GR, MDIM, NDIM, NDIM, nullptr, 0, stream);

    // ---- Z = (GR @ H) @ GT ----
    float* Y = (GR == bufC) ? bufD : bufC;
    launch_cgemm(GR, H, Y, MDIM, NDIM, NDIM, nullptr, 0, stream);
    launch_cgemm(Y, GT, out, MDIM, NDIM, MDIM, nullptr, 0, stream);
}